// QNN_14654428414825
// MI455X (gfx1250) — compile-verified
//
#include <hip/hip_runtime.h>

// ---------------------------------------------------------------------------
// QNN forward on MI455X (gfx1250, wave32, WMMA + TDM async tensor loads).
//   out = ((Xnorm @ R) ** 2) @ W^T + b,   R = circuit matrix transpose (1024x1024)
// ---------------------------------------------------------------------------

typedef __attribute__((ext_vector_type(2))) float v2f;
typedef __attribute__((ext_vector_type(8))) float v8f;
typedef __attribute__((ext_vector_type(4))) unsigned int v4u;
typedef __attribute__((ext_vector_type(8))) int v8i;
typedef __attribute__((ext_vector_type(4))) int v4i;

#define NQ      10
#define DIM     1024
#define BATCH   16384
#define IMG     784        // 28*28
#define BM      128
#define BN      128
#define NBN     (DIM / BN) // 8
#define LDA     34         // A LDS row stride (floats), bank-spread
#define LDB     136        // B LDS row stride (floats); TDM pad reproduces this
#define LDP     132        // P LDS row stride (floats)

// dynamic LDS layout (floats):
//   As[2] : 2 * 128*34 = 8704     @ 0
//   Bs[2] : 2 * 32*136 = 8704     @ 8704
//   Wt    : 128*16     = 2048     @ 17408
//   Ps    : 64*132     = 8448     overlays As (after main loop)
#define ASZ   (BM * LDA)           // 4352
#define BSZ   (32 * LDB)           // 4352
#define SMEM_FLOATS (2 * ASZ + 2 * BSZ + 2048)   // 19456
#define SMEM_BYTES  (SMEM_FLOATS * 4)            // 77824

// ---------------------------------------------------------------------------
// Kernel 1: build R = M^T by running the circuit on identity rows.
// ---------------------------------------------------------------------------
__global__ __launch_bounds__(128) void build_R(const float* __restrict__ angles,
                                               float* __restrict__ R) {
  __shared__ float st[DIM];
  const int brow = blockIdx.x;
  const int t = threadIdx.x;
  for (int i = t; i < DIM; i += 128) st[i] = (i == brow) ? 1.0f : 0.0f;
  __syncthreads();
  for (int q = 0; q < NQ; ++q) {           // Ry layer: qubit q <-> bit 9-q
    const float ang = angles[q];           // N_LAYERS == 1
    const float c = cosf(0.5f * ang);
    const float s = sinf(0.5f * ang);
    const int bit = 1 << (9 - q);
    for (int p = t; p < DIM / 2; p += 128) {
      const int low = p & (bit - 1);
      const int i0 = ((p ^ low) << 1) | low;
      const int i1 = i0 | bit;
      const float lo = st[i0], hi = st[i1];
      st[i0] = c * lo - s * hi;
      st[i1] = s * lo + c * hi;
    }
    __syncthreads();
  }
  for (int q = 0; q < NQ - 1; ++q) {       // CX chain: swap tgt where ctrl=1
    const int cb = 1 << (9 - q);
    const int tb = 1 << (8 - q);
    for (int p = t; p < DIM / 4; p += 128) {
      const int low = p & (tb - 1);
      const int i0 = ((p ^ low) << 2) | cb | low;
      const int i1 = i0 | tb;
      const float a = st[i0], bswap = st[i1];
      st[i0] = bswap;
      st[i1] = a;
    }
    __syncthreads();
  }
  for (int i = t; i < DIM; i += 128) R[brow * DIM + i] = st[i];
}

// ---------------------------------------------------------------------------
// Kernel 2: per-row inverse L2 norms of x. One wave32 per row.
// ---------------------------------------------------------------------------
__global__ __launch_bounds__(256) void row_norms(const float* __restrict__ x,
                                                 float* __restrict__ invnorm) {
  const int row = blockIdx.x * 8 + (threadIdx.x >> 5);
  const int lane = threadIdx.x & 31;
  const float* p = x + (size_t)row * IMG;
  float s = 0.0f;
  for (int i = lane; i < IMG; i += 32) { const float v = p[i]; s += v * v; }
#pragma unroll
  for (int off = 16; off > 0; off >>= 1) s += __shfl_xor(s, off, 32);
  if (lane == 0) invnorm[row] = rsqrtf(s);
}

// ---------------------------------------------------------------------------
// TDM: async-load one 32x128 f32 tile of R (row stride 1024) into LDS with
// +8-DWORD padding every 128 DWORDs (-> LDB=136 layout). 6-arg builtin form
// (this toolchain): (v4u g0, v8i g1, v4i, v4i, v8i, i32 cpol).
// D# bitfields per CDNA5 ISA.
// ---------------------------------------------------------------------------
__device__ __forceinline__ void tdm_load_B(const float* gsrc, unsigned lds_byte) {
  const unsigned long long ga = (unsigned long long)(uintptr_t)gsrc;
  v4u g0;
  g0[0] = 1u;                                           // count=1, user mode
  g0[1] = lds_byte;                                     // lds_addr (bytes)
  g0[2] = (unsigned)ga;                                 // global_addr[31:0]
  g0[3] = (unsigned)((ga >> 32) & 0x01FFFFFFu)          // global_addr[56:32]
          | (2u << 30);                                 // type=2 ("image")
  v8i g1;
  g1[0] = (int)((2u << 16)      // data_size = 4B
              | (1u << 20)      // pad_enable
              | (6u << 22)      // pad_interval: 2^(6+1)=128 DWORDs
              | (7u << 25));    // pad_amount: 7 -> 8 DWORDs
  g1[1] = (int)(1024u << 16);   // tensor_dim0 = 1024 (low16 @ [31:16])
  g1[2] = (int)(1024u << 16);   // dim0 hi=0 | tensor_dim1 low16 = 1024
  g1[3] = (int)(128u << 16);    // dim1 hi=0 | tile_dim0 = 128
  g1[4] = 32;                   // tile_dim1 = 32, tile_dim2 = 0
  g1[5] = 1024;                 // tensor_dim0_stride low32 (elements)
  g1[6] = 0;                    // stride hi | tensor_dim1_stride low16
  g1[7] = 0;
  const v4i z4 = {0, 0, 0, 0};              // groups 2/3 unused (2D tile)
  const v8i z8 = {0, 0, 0, 0, 0, 0, 0, 0};  // extra group slot unused
  __builtin_amdgcn_tensor_load_to_lds(g0, g1, z4, z4, z8, 0);
}

// ---------------------------------------------------------------------------
// Kernel 3: main fused GEMM, double-buffered, TDM for B, pipelined A loads.
// ---------------------------------------------------------------------------
__global__ __launch_bounds__(256) void qnn_gemm(const float* __restrict__ x,
                                                const float* __restrict__ R,
                                                const float* __restrict__ W,
                                                const float* __restrict__ inorm,
                                                float* __restrict__ partial) {
  extern __shared__ float smem[];
  float* const Wt = smem + 2 * ASZ + 2 * BSZ;   // [128][16], t>=10 zero
  float* const Ps = smem;                       // [64][LDP] overlay (epilogue)

  const int bm = blockIdx.x;             // 0..127  (M block)
  const int bn = blockIdx.y;             // 0..7    (N block)
  const int tid = threadIdx.x;
  const int lane = tid & 31;
  const int w = tid >> 5;                // wave id 0..7
  const int half = lane >> 4;            // 0/1
  const int idx = lane & 15;
  const int wm = w & 3;                  // wave M coord (4)
  const int wn = w >> 2;                 // wave N coord (2)

  // Stage W band -> Wt[k][t]
#pragma unroll
  for (int i = 0; i < 8; ++i) {
    const int id = tid + 256 * i;
    const int kk = id >> 4, tt = id & 15;
    Wt[id] = (tt < 10) ? W[tt * DIM + bn * BN + kk] : 0.0f;
  }

  const v8f vzero = {0, 0, 0, 0, 0, 0, 0, 0};
  v8f acc[2][4];
#pragma unroll
  for (int sm = 0; sm < 2; ++sm)
#pragma unroll
    for (int sn = 0; sn < 4; ++sn) acc[sm][sn] = vzero;

  const int colA = tid & 31;             // padded-image column 0..31
  const int mrow0 = tid >> 5;            // 0..7
  const bool cin = (colA >= 2) && (colA < 30);

  // Hoist per-row inverse norms (independent of r)
  float inr[16];
#pragma unroll
  for (int i = 0; i < 16; ++i) inr[i] = inorm[bm * BM + mrow0 + 8 * i];

  // ---- Prologue: prime buffer 0 with r = 2 ----
  if (w == 0) tdm_load_B(R + (2 * 32) * DIM + bn * BN,
                         (unsigned)(uintptr_t)(smem + 2 * ASZ));
  {
    float* const A0 = smem;
#pragma unroll
    for (int i = 0; i < 16; ++i) {
      const int m = mrow0 + 8 * i;
      float v = 0.0f;
      if (cin) v = x[(bm * BM + m) * IMG + 0 * 28 + (colA - 2)];
      A0[m * LDA + colA] = v * inr[i];
    }
  }

  // ---- Main loop: r = 2..29 (nonzero padded rows), double buffered ----
  for (int r = 2; r < 30; ++r) {
    const int cur = (r - 2) & 1, nxt = cur ^ 1;
    float* const Acur = smem + cur * ASZ;
    float* const Bcur = smem + 2 * ASZ + cur * BSZ;
    float* const Anxt = smem + nxt * ASZ;

    if (w == 0) __builtin_amdgcn_s_wait_tensorcnt(0);  // B[cur] landed in LDS
    __syncthreads();   // B[cur] visible; A[cur] stores visible; prev compute done

    const bool have_next = (r + 1 < 30);   // uniform
    float areg[16];
    if (have_next) {
      if (w == 0) tdm_load_B(R + ((r + 1) * 32) * DIM + bn * BN,
                             (unsigned)(uintptr_t)(smem + 2 * ASZ + nxt * BSZ));
#pragma unroll
      for (int i = 0; i < 16; ++i) {       // issue global loads early
        const int m = mrow0 + 8 * i;
        areg[i] = cin ? x[(bm * BM + m) * IMG + (r - 1) * 28 + (colA - 2)] : 0.0f;
      }
    }

    // 8 k-steps of 4: 64 wmma per wave per K-tile (hides the loads above)
#pragma unroll
    for (int ks = 0; ks < 8; ++ks) {
      const int kb = 4 * ks;
      v2f a[2], bf[4];
#pragma unroll
      for (int sm = 0; sm < 2; ++sm) {
        const int mrow = wm * 32 + sm * 16 + idx;
        a[sm].x = Acur[mrow * LDA + kb + 2 * half];
        a[sm].y = Acur[mrow * LDA + kb + 2 * half + 1];
      }
#pragma unroll
      for (int sn = 0; sn < 4; ++sn) {
        const int ncol = wn * 64 + sn * 16 + idx;
        bf[sn].x = Bcur[(kb + 2 * half) * LDB + ncol];
        bf[sn].y = Bcur[(kb + 2 * half + 1) * LDB + ncol];
      }
#pragma unroll
      for (int sm = 0; sm < 2; ++sm)
#pragma unroll
        for (int sn = 0; sn < 4; ++sn)
          acc[sm][sn] = __builtin_amdgcn_wmma_f32_16x16x4_f32(
              false, a[sm], false, bf[sn], (short)0, acc[sm][sn], false, false);
    }

    if (have_next) {                       // stash next A tile (waits loadcnt here)
#pragma unroll
      for (int i = 0; i < 16; ++i) {
        const int m = mrow0 + 8 * i;
        Anxt[m * LDA + colA] = areg[i] * inr[i];
      }
    }
  }

  // ---- Epilogue: out_part = (Y^2) @ Wt, two 64-row phases through Ps.
  // Wave-uniform branches -> EXEC stays all-ones for WMMA.
  v8f oacc = vzero;

  __syncthreads();
  if (wm < 2) {                            // rows 0..63 write squares
#pragma unroll
    for (int sm = 0; sm < 2; ++sm)
#pragma unroll
      for (int sn = 0; sn < 4; ++sn)
#pragma unroll
        for (int v = 0; v < 8; ++v) {
          const int m = wm * 32 + sm * 16 + v + 8 * half;
          const int n = wn * 64 + sn * 16 + idx;
          const float y = acc[sm][sn][v];
          Ps[m * LDP + n] = y * y;
        }
  }
  __syncthreads();
  if (w < 4) {                             // rows 16w..16w+15, K = 128
#pragma unroll
    for (int kb = 0; kb < 128; kb += 4) {
      v2f a, bf;
      const int mr = 16 * w + idx;
      a.x = Ps[mr * LDP + kb + 2 * half];
      a.y = Ps[mr * LDP + kb + 2 * half + 1];
      bf.x = Wt[(kb + 2 * half) * 16 + idx];
      bf.y = Wt[(kb + 2 * half + 1) * 16 + idx];
      oacc = __builtin_amdgcn_wmma_f32_16x16x4_f32(false, a, false, bf,
                                                   (short)0, oacc, false, false);
    }
  }
  __syncthreads();
  if (wm >= 2) {                           // rows 64..127 write squares
#pragma unroll
    for (int sm = 0; sm < 2; ++sm)
#pragma unroll
      for (int sn = 0; sn < 4; ++sn)
#pragma unroll
        for (int v = 0; v < 8; ++v) {
          const int m = wm * 32 + sm * 16 + v + 8 * half - 64;
          const int n = wn * 64 + sn * 16 + idx;
          const float y = acc[sm][sn][v];
          Ps[m * LDP + n] = y * y;
        }
  }
  __syncthreads();
  if (w >= 4) {                            // global rows 16w
#pragma unroll
    for (int kb = 0; kb < 128; kb += 4) {
      v2f a, bf;
      const int mr = 16 * (w - 4) + idx;
      a.x = Ps[mr * LDP + kb + 2 * half];
      a.y = Ps[mr * LDP + kb + 2 * half + 1];
      bf.x = Wt[(kb + 2 * half) * 16 + idx];
      bf.y = Wt[(kb + 2 * half + 1) * 16 + idx];
      oacc = __builtin_amdgcn_wmma_f32_16x16x4_f32(false, a, false, bf,
                                                   (short)0, oacc, false, false);
    }
  }

  // Store deterministic partial: [bm][bn][m_local 128][t 16]
  {
    const int base = ((bm * NBN + bn) * BM + 16 * w) * 16;
#pragma unroll
    for (int v = 0; v < 8; ++v) {
      const int m = v + 8 * half;
      partial[base + m * 16 + idx] = oacc[v];
    }
  }
}

// ---------------------------------------------------------------------------
// Kernel 4: deterministic reduction over the 8 N-band partials + bias.
// ---------------------------------------------------------------------------
__global__ __launch_bounds__(256) void reduce_out(const float* __restrict__ partial,
                                                  const float* __restrict__ bias,
                                                  float* __restrict__ out) {
  const int i = blockIdx.x * 256 + threadIdx.x;
  if (i >= BATCH * 10) return;
  const int m = i / 10, t = i % 10;
  const int bmb = m >> 7, ml = m & 127;
  float s = bias[t];
#pragma unroll
  for (int bn = 0; bn < NBN; ++bn)
    s += partial[((bmb * NBN + bn) * BM + ml) * 16 + t];
  out[i] = s;
}

// ---------------------------------------------------------------------------
extern "C" void kernel_launch(void* const* d_in, const int* in_sizes, int n_in,
                              void* d_out, int out_size, void* d_ws, size_t ws_size,
                              hipStream_t stream) {
  (void)in_sizes; (void)n_in; (void)out_size; (void)ws_size;
  const float* x      = (const float*)d_in[0];  // (16384,1,28,28)
  const float* angles = (const float*)d_in[1];  // (1,10)
  const float* W      = (const float*)d_in[2];  // (10,1024)
  const float* bias   = (const float*)d_in[3];  // (10,)
  float* out = (float*)d_out;                   // (16384,10)

  char* ws = (char*)d_ws;
  float* R       = (float*)ws;                                      // 4 MB
  float* invnorm = (float*)(ws + (size_t)(4u << 20));               // 64 KB
  float* partial = (float*)(ws + (size_t)(4u << 20) + (64u << 10)); // 8 MB

  build_R<<<DIM, 128, 0, stream>>>(angles, R);
  row_norms<<<BATCH / 8, 256, 0, stream>>>(x, invnorm);
  qnn_gemm<<<dim3(BATCH / BM, NBN), 256, SMEM_BYTES, stream>>>(x, R, W, invnorm,
                                                               partial);
  reduce_out<<<(BATCH * 10 + 255) / 256, 256, 0, stream>>>(partial, bias, out);
}